// Low_Rank_RNN_Dynamics_54314156425438
// MI455X (gfx1250) — compile-verified
//
#include <hip/hip_runtime.h>
#include <hip/hip_bf16.h>

#define B_DIM   128
#define T_DIM   1000
#define IN_DIM  3
#define OUT_DIM 3
#define HID     512
#define NOISE_STD 0.05f
#define TAU       0.2f

// gfx1250 hardware tanh if the builtin exists (v_tanh_f32), else libm fallback
#if defined(__has_builtin)
#  if __has_builtin(__builtin_amdgcn_tanhf)
#    define TANHF(x) __builtin_amdgcn_tanhf((x))
#  elif __has_builtin(__builtin_amdgcn_tanh_f32)
#    define TANHF(x) __builtin_amdgcn_tanh_f32((x))
#  endif
#endif
#ifndef TANHF
#  define TANHF(x) tanhf((x))
#endif

typedef float v2f __attribute__((ext_vector_type(2)));
typedef float v8f __attribute__((ext_vector_type(8)));

// ---------------------------------------------------------------------------
// Kernel A: the sequential scan. One block per batch item, one hidden unit
// per thread. Critical path per step: rank-2 wave shuffle-reduce + one
// barrier (double-buffered LDS partials) + hardware tanh. Noise is
// prefetched 2 steps ahead; u (uniform, SMEM) 1 step ahead.
// ---------------------------------------------------------------------------
__global__ __launch_bounds__(HID, 1)
void rnn_scan_kernel(const float* __restrict__ u,      // [B,T,IN]
                     const float* __restrict__ x0,     // [B,HID]
                     const float* __restrict__ noise,  // [T,B,HID]
                     const float* __restrict__ M,      // [HID,2]
                     const float* __restrict__ N,      // [HID,2]
                     const float* __restrict__ bias,   // [HID]
                     const float* __restrict__ Win,    // [HID,IN]
                     float* __restrict__ x_final,      // [B,HID]
                     float* __restrict__ traj)         // [B,T+1,HID]
{
    const int b    = blockIdx.x;
    const int h    = threadIdx.x;          // 0..511
    const int lane = h & 31;
    const int wv   = h >> 5;               // wave index 0..15

    __shared__ float2 part[2][HID / 32];   // double-buffered cross-wave partials

    // Register-resident per-hidden-unit weights
    const float m0 = M[h * 2 + 0], m1 = M[h * 2 + 1];
    const float n0 = N[h * 2 + 0], n1 = N[h * 2 + 1];
    const float bi = bias[h];
    const float w0 = Win[h * 3 + 0], w1 = Win[h * 3 + 1], w2 = Win[h * 3 + 2];

    float* trow = traj + (size_t)b * (T_DIM + 1) * HID;
    const float* nzbase = noise + (size_t)b * HID + h;
    const float* ub     = u + (size_t)b * T_DIM * IN_DIM;

    float x = x0[(size_t)b * HID + h];
    trow[h] = x;                           // trajectories[:,0,:] = x0

    // 2-deep noise pipeline: t=0 and t=1 in flight before the scan starts
    float nzA = nzbase[0];
    float nzB = nzbase[(size_t)B_DIM * HID];

    // u for t=0 (uniform address -> scalar loads)
    float u0 = ub[0], u1 = ub[1], u2 = ub[2];

    // r0 = tanh(x0)   (no bias, per reference)
    float r = TANHF(x);

    // initial kappa reduction (buffer 0)
    float p0 = r * n0, p1 = r * n1;
#pragma unroll
    for (int off = 16; off > 0; off >>= 1) {
        p0 += __shfl_xor(p0, off, 32);
        p1 += __shfl_xor(p1, off, 32);
    }
    if (lane == 0) part[0][wv] = make_float2(p0, p1);
    __syncthreads();
    float k0 = 0.f, k1 = 0.f;
#pragma unroll
    for (int w = 0; w < HID / 32; ++w) { float2 q = part[0][w]; k0 += q.x; k1 += q.y; }

    for (int t = 0; t < T_DIM; ++t) {
        const int buf = (t + 1) & 1;

        // input projection for this step
        const float I = u0 * w0 + u1 * w1 + u2 * w2;

        // prefetch next step's u (uniform)
        if (t + 1 < T_DIM) {
            const float* un = ub + (size_t)(t + 1) * IN_DIM;
            u0 = un[0]; u1 = un[1]; u2 = un[2];
        }

        // low-rank recurrent drive
        const float hw = (k0 * m0 + k1 * m1) * (1.0f / (float)HID);

        // state update
        x = x + NOISE_STD * nzA + TAU * (-x + hw + I);

        // rotate noise pipeline; issue load for t+2
        nzA = nzB;
        if (t + 2 < T_DIM)
            nzB = nzbase[(size_t)(t + 2) * B_DIM * HID];

        // coalesced trajectory write
        trow[(size_t)(t + 1) * HID + h] = x;

        r  = TANHF(x + bi);
        p0 = r * n0;
        p1 = r * n1;
#pragma unroll
        for (int off = 16; off > 0; off >>= 1) {
            p0 += __shfl_xor(p0, off, 32);
            p1 += __shfl_xor(p1, off, 32);
        }
        if (lane == 0) part[buf][wv] = make_float2(p0, p1);
        __syncthreads();
        k0 = 0.f; k1 = 0.f;
#pragma unroll
        for (int w = 0; w < HID / 32; ++w) { float2 q = part[buf][w]; k0 += q.x; k1 += q.y; }
    }

    x_final[(size_t)b * HID + h] = x;
}

// ---------------------------------------------------------------------------
// Kernel B: output projection out[b,t,o] = sum_h tanh(traj[b,t+1,h])*Wout[o,h]
// via V_WMMA_F32_16X16X4_F32. One 16-row tile per wave; 128 K-chunks of 4.
// A fragment (16x4 f32): lane[3:0]=M row, lane[4] selects K pair {0,1}/{2,3}.
// B fragments (4x16 f32, from Wout, zero-padded for n>=3) are precomputed
// once per block into LDS so the hot loop has no divergence:
//   global b64 (A) -> 2x v_tanh_f32 -> ds b64 (B) -> v_wmma.
// C/D (16x16 f32): vgpr j -> row (lane[4]? j+8 : j), col = lane[3:0].
// ---------------------------------------------------------------------------
#define ROWS_TOTAL (B_DIM * T_DIM)     /* 128000, divisible by 16 */
#define TILES      (ROWS_TOTAL / 16)   /* 8000 */
#define WAVES_PB   8
#define KCHUNKS    (HID / 4)           /* 128 */

__global__ __launch_bounds__(32 * WAVES_PB, 1)
void out_proj_wmma_kernel(const float* __restrict__ traj,  // [B,T+1,HID]
                          const float* __restrict__ Wout,  // [OUT,HID]
                          float* __restrict__ out)         // [B,T,OUT]
{
    const int lane  = threadIdx.x & 31;
    const int wv    = threadIdx.x >> 5;
    const int tile  = blockIdx.x * WAVES_PB + wv;
    const int rlow  = lane & 15;            // M row (A) / N col (B)
    const int khalf = (lane >> 4) << 1;     // 0 or 2 within each K-chunk of 4

    // Precompute all Wout B-fragments into LDS: [chunk][lane] -> float2
    __shared__ float2 bfrag[KCHUNKS][32];   // 32 KB
    for (int c = wv; c < KCHUNKS; c += WAVES_PB) {
        const int kk = 4 * c + khalf;
        float2 v = make_float2(0.f, 0.f);
        if (rlow < OUT_DIM) {
            v.x = Wout[(size_t)rlow * HID + kk];
            v.y = Wout[(size_t)rlow * HID + kk + 1];
        }
        bfrag[c][lane] = v;
    }
    __syncthreads();

    const int  row = tile * 16 + rlow;      // row in [0, 128000)
    const int  b   = row / T_DIM;
    const int  t   = row - b * T_DIM;
    const float* arow = traj + ((size_t)b * (T_DIM + 1) + t + 1) * HID + khalf;

    v8f acc = {0.f, 0.f, 0.f, 0.f, 0.f, 0.f, 0.f, 0.f};

#if defined(__has_builtin) && __has_builtin(__builtin_amdgcn_wmma_f32_16x16x4_f32)
#pragma unroll 8
    for (int c = 0; c < KCHUNKS; ++c) {
        const v2f a2 = *(const v2f*)(arow + 4 * c);   // 8B-aligned b64 load
        v2f a;
        a.x = TANHF(a2.x);
        a.y = TANHF(a2.y);
        const float2 bw = bfrag[c][lane];
        v2f bb = {bw.x, bw.y};
        acc = __builtin_amdgcn_wmma_f32_16x16x4_f32(
            /*neg_a=*/false, a, /*neg_b=*/false, bb,
            /*c_mod=*/(short)0, acc, /*reuse_a=*/false, /*reuse_b=*/false);
    }
#else
    // Scalar fallback (should not be taken on gfx1250)
    for (int c = 0; c < KCHUNKS; ++c) {
        const v2f a2 = *(const v2f*)(arow + 4 * c);
        const float2 bw = bfrag[c][lane];
        acc[0] += TANHF(a2.x) * bw.x + TANHF(a2.y) * bw.y;
    }
#endif

    // Scatter the 3 useful columns of the 16x16 D tile
    const int ncol  = lane & 15;
    const int mbase = (lane >> 4) * 8;
    if (ncol < OUT_DIM) {
#pragma unroll
        for (int j = 0; j < 8; ++j) {
            const int grow = tile * 16 + mbase + j;       // (b*T + t) flat row
            out[(size_t)grow * OUT_DIM + ncol] = acc[j];
        }
    }
}

// ---------------------------------------------------------------------------
extern "C" void kernel_launch(void* const* d_in, const int* in_sizes, int n_in,
                              void* d_out, int out_size, void* d_ws, size_t ws_size,
                              hipStream_t stream) {
    (void)in_sizes; (void)n_in; (void)d_ws; (void)ws_size; (void)out_size;

    const float* u     = (const float*)d_in[0];   // [B,T,IN]
    const float* x0    = (const float*)d_in[1];   // [B,HID]
    const float* noise = (const float*)d_in[2];   // [T,B,HID]
    const float* M     = (const float*)d_in[3];   // [HID,2]
    const float* N     = (const float*)d_in[4];   // [HID,2]
    const float* bias  = (const float*)d_in[5];   // [HID]
    const float* Win   = (const float*)d_in[6];   // [HID,IN]
    const float* Wout  = (const float*)d_in[7];   // [OUT,HID]

    float* out     = (float*)d_out;                               // [B,T,OUT]
    float* x_final = out + (size_t)B_DIM * T_DIM * OUT_DIM;       // [B,HID]
    float* traj    = x_final + (size_t)B_DIM * HID;               // [B,T+1,HID]

    rnn_scan_kernel<<<B_DIM, HID, 0, stream>>>(u, x0, noise, M, N, bias, Win,
                                               x_final, traj);

    out_proj_wmma_kernel<<<TILES / WAVES_PB, 32 * WAVES_PB, 0, stream>>>(
        traj, Wout, out);
}